// VectorQuantizer_45827301048597
// MI455X (gfx1250) — compile-verified
//
#include <hip/hip_runtime.h>

#define Bdim 64
#define Sdim 512
#define Ddim 128
#define Kcb  1024
#define Ntok (Bdim * Sdim)          // 32768
#define NElem ((size_t)Ntok * Ddim) // 4194304

typedef float v2f __attribute__((ext_vector_type(2)));
typedef float v8f __attribute__((ext_vector_type(8)));

// ---------------------------------------------------------------------------
// Phase 0: per-code squared norms; zero counts + SSE accumulator.
// ---------------------------------------------------------------------------
__global__ void vq_init(const float* __restrict__ emb, float* __restrict__ enorm,
                        float* __restrict__ counts, float* __restrict__ sse) {
    int t = blockIdx.x * blockDim.x + threadIdx.x;
    if (t < Kcb) {
        const float* e = emb + (size_t)t * Ddim;
        float s = 0.f;
#pragma unroll 8
        for (int d = 0; d < Ddim; ++d) s += e[d] * e[d];
        enorm[t]  = s;
        counts[t] = 0.f;
    }
    if (t == 0) *sse = 0.f;
}

// ---------------------------------------------------------------------------
// Phase 1: fused distance GEMM + argmin.
// Block = 256 threads (8 waves). Wave w owns TWO 16-token tiles (32 tokens),
// so each LDS B-fragment feeds 2 WMMAs across 2 independent accumulator
// chains. Block covers 256 tokens; grid = Ntok/256 = 128.
// emb staged in LDS, 16 codes/block, row stride 132 floats (bank-conflict-free
// b64 reads: lane banks = 4*(lane&15) + 2*(lane>>4) + {0,1} -> 64 distinct).
// ---------------------------------------------------------------------------
__launch_bounds__(256)
__global__ void vq_argmin(const float* __restrict__ z, const float* __restrict__ emb,
                          const float* __restrict__ enorm_g, int* __restrict__ idx) {
    __shared__ float lds_e[16 * 132];
    __shared__ float lds_n[Kcb];

    const int tid = threadIdx.x;
    for (int i = tid; i < Kcb; i += 256) lds_n[i] = enorm_g[i];

    const int lane = tid & 31;
    const int wv   = tid >> 5;
    const int n    = lane & 15;   // B/C column, code-within-block
    const int h    = lane >> 4;   // half-wave select
    const int tokBase = blockIdx.x * 256 + wv * 32;

    // Preload A fragments for both 16xD z-tiles (32 k-steps of 4 each).
    v2f a0[32], a1[32];
    const float* zp0 = z + (size_t)(tokBase + n) * Ddim + 2 * h;
    const float* zp1 = zp0 + 16 * Ddim;
#pragma unroll
    for (int kk = 0; kk < 32; ++kk) {
        a0[kk] = *(const v2f*)(zp0 + 4 * kk);
        a1[kk] = *(const v2f*)(zp1 + 4 * kk);
    }

    float best0[8], best1[8];
    int   bidx0[8], bidx1[8];
#pragma unroll
    for (int j = 0; j < 8; ++j) {
        best0[j] = 3.0e38f; bidx0[j] = 0;
        best1[j] = 3.0e38f; bidx1[j] = 0;
    }

    // Cooperative emb-block staging: 16 codes x 128 dims = 2048 floats/block.
    const int e0 = tid * 4, e1 = tid * 4 + 1024;
    float4 g0 = *(const float4*)(emb + (size_t)(e0 >> 7) * Ddim + (e0 & 127));
    float4 g1 = *(const float4*)(emb + (size_t)(e1 >> 7) * Ddim + (e1 & 127));

    for (int cb = 0; cb < Kcb; cb += 16) {
        *(float4*)&lds_e[(e0 >> 7) * 132 + (e0 & 127)] = g0;
        *(float4*)&lds_e[(e1 >> 7) * 132 + (e1 & 127)] = g1;
        if (cb + 16 < Kcb) {  // prefetch next block; consumed after compute
            g0 = *(const float4*)(emb + (size_t)(cb + 16 + (e0 >> 7)) * Ddim + (e0 & 127));
            g1 = *(const float4*)(emb + (size_t)(cb + 16 + (e1 >> 7)) * Ddim + (e1 & 127));
        }
        __syncthreads();

        v8f acc0 = {}, acc1 = {};
        const float* bp = &lds_e[n * 132 + 2 * h];
#pragma unroll
        for (int kk = 0; kk < 32; ++kk) {
            v2f b = *(const v2f*)(bp + 4 * kk);
            acc0 = __builtin_amdgcn_wmma_f32_16x16x4_f32(
                false, a0[kk], false, b, (short)0, acc0, false, false);
            acc1 = __builtin_amdgcn_wmma_f32_16x16x4_f32(
                false, a1[kk], false, b, (short)0, acc1, false, false);
        }

        const float en = lds_n[cb + n];  // ||e||^2 for this lane's code
#pragma unroll
        for (int j = 0; j < 8; ++j) {
            float v0 = en - 2.0f * acc0[j];  // dist - ||z||^2 (row-constant)
            if (v0 < best0[j]) { best0[j] = v0; bidx0[j] = cb + n; }
            float v1 = en - 2.0f * acc1[j];
            if (v1 < best1[j]) { best1[j] = v1; bidx1[j] = cb + n; }
        }
        __syncthreads();  // protect lds_e before next iteration's store
    }

    // Min+index reduce across the 16 lanes holding one C row (same h group).
#pragma unroll
    for (int j = 0; j < 8; ++j) {
        float bv0 = best0[j]; int bi0 = bidx0[j];
        float bv1 = best1[j]; int bi1 = bidx1[j];
#pragma unroll
        for (int m = 8; m >= 1; m >>= 1) {
            float ov0 = __shfl_xor(bv0, m, 32);
            int   oi0 = __shfl_xor(bi0, m, 32);
            if (ov0 < bv0 || (ov0 == bv0 && oi0 < bi0)) { bv0 = ov0; bi0 = oi0; }
            float ov1 = __shfl_xor(bv1, m, 32);
            int   oi1 = __shfl_xor(bi1, m, 32);
            if (ov1 < bv1 || (ov1 == bv1 && oi1 < bi1)) { bv1 = ov1; bi1 = oi1; }
        }
        if (n == 0) {
            idx[tokBase + j + 8 * h]      = bi0;  // C row M = j + 8*h, tile 0
            idx[tokBase + 16 + j + 8 * h] = bi1;  // tile 1
        }
    }
}

// ---------------------------------------------------------------------------
// Phase 2: gather z_q, straight-through output, SSE accumulation.
// ---------------------------------------------------------------------------
__global__ void vq_gather(const float* __restrict__ z, const float* __restrict__ emb,
                          const int* __restrict__ idx, float* __restrict__ outq,
                          float* __restrict__ sse) {
    __shared__ float red[256];
    size_t e = (size_t)blockIdx.x * 256 + threadIdx.x;
    float local = 0.f;
    if (e < NElem) {
        int token = (int)(e >> 7);
        int d     = (int)(e & 127);
        int code  = idx[token];
        float zq = emb[(size_t)code * Ddim + d];
        float zv = z[e];
        outq[e] = zv + (zq - zv);  // matches reference STE arithmetic bitwise
        float diff = zq - zv;
        local = diff * diff;
    }
    red[threadIdx.x] = local;
    __syncthreads();
    for (int s = 128; s > 0; s >>= 1) {
        if (threadIdx.x < s) red[threadIdx.x] += red[threadIdx.x + s];
        __syncthreads();
    }
    if (threadIdx.x == 0) unsafeAtomicAdd(sse, red[0]);
}

// ---------------------------------------------------------------------------
// Phase 3: histogram of code usage over valid tokens.
// ---------------------------------------------------------------------------
__global__ void vq_counts(const unsigned char* __restrict__ mask,
                          const int* __restrict__ idx, float* __restrict__ counts) {
    int t = blockIdx.x * 256 + threadIdx.x;
    if (t < Ntok && mask[t] == 0) unsafeAtomicAdd(&counts[idx[t]], 1.0f);
}

// ---------------------------------------------------------------------------
// Phase 4: loss + perplexity (single block, K threads).
// ---------------------------------------------------------------------------
__launch_bounds__(1024)
__global__ void vq_final(const float* __restrict__ counts, const float* __restrict__ sse,
                         float* __restrict__ out_scalars) {
    __shared__ float red[1024];
    int t = threadIdx.x;
    float c = counts[t];
    red[t] = c;
    __syncthreads();
    for (int s = 512; s > 0; s >>= 1) {
        if (t < s) red[t] += red[t + s];
        __syncthreads();
    }
    float total = red[0];
    __syncthreads();
    float p = c / total;
    red[t] = p * logf(p + 1e-10f);
    __syncthreads();
    for (int s = 512; s > 0; s >>= 1) {
        if (t < s) red[t] += red[t + s];
        __syncthreads();
    }
    if (t == 0) {
        out_scalars[0] = 1.25f * (*sse) / (float)NElem;  // q_loss + 0.25*e_loss
        out_scalars[1] = expf(-red[0]);                  // perplexity
    }
}

// ---------------------------------------------------------------------------
extern "C" void kernel_launch(void* const* d_in, const int* in_sizes, int n_in,
                              void* d_out, int out_size, void* d_ws, size_t ws_size,
                              hipStream_t stream) {
    const float*         z    = (const float*)d_in[0];
    const float*         emb  = (const float*)d_in[1];
    const unsigned char* mask = (const unsigned char*)d_in[2];
    float* out = (float*)d_out;

    // ws layout (floats): [0,K) enorm | [K,2K) counts | [2K] sse | then N ints idx
    float* ws     = (float*)d_ws;
    float* enorm  = ws;
    float* counts = ws + Kcb;
    float* sse    = ws + 2 * Kcb;
    int*   idx    = (int*)(ws + 2 * Kcb + 1);

    vq_init  <<<(Kcb + 255) / 256, 256, 0, stream>>>(emb, enorm, counts, sse);
    vq_argmin<<<Ntok / 256,        256, 0, stream>>>(z, emb, enorm, idx);
    vq_gather<<<(int)((NElem + 255) / 256), 256, 0, stream>>>(z, emb, idx, out, sse);
    vq_counts<<<(Ntok + 255) / 256, 256, 0, stream>>>(mask, idx, counts);
    vq_final <<<1, 1024, 0, stream>>>(counts, sse, out + NElem);
}